// TmrtGAT_5738076307491
// MI455X (gfx1250) — compile-verified
//
#include <hip/hip_runtime.h>
#include <hip/hip_bf16.h>

typedef float v2f __attribute__((ext_vector_type(2)));
typedef float v8f __attribute__((ext_vector_type(8)));

#define NEG_SLOPE 0.2f
#define F 128   // HEADS*HID
#define H4 4    // heads
#define C32 32  // channels per head

// ---------------------------------------------------------------- fill
__global__ void gat_fill_u32(unsigned* __restrict__ p, unsigned v, size_t n) {
    size_t tid = (size_t)blockIdx.x * blockDim.x + threadIdx.x;
    if (tid < n) p[tid] = v;
}

// ---------------------------------------------------------------- layer1 projection: h = x @ W1  ([N,8]@[8,128])
__global__ __launch_bounds__(256) void gat_lin1(const float* __restrict__ x,
                                                const float* __restrict__ W1,
                                                float* __restrict__ h, int N) {
    size_t tid = (size_t)blockIdx.x * 256 + threadIdx.x;
    if (tid >= (size_t)N * F) return;
    int n = (int)(tid >> 7);
    int j = (int)(tid & (F - 1));
    const float* xr = x + (size_t)n * 8;
    float s = 0.f;
#pragma unroll
    for (int k = 0; k < 8; ++k) s = fmaf(xr[k], W1[k * F + j], s);
    h[tid] = s;
}

// ---------------------------------------------------------------- attention coefficients a_src/a_dst per (node, head)
__global__ __launch_bounds__(256) void gat_att(const float* __restrict__ h,
                                               const float* __restrict__ att_s,
                                               const float* __restrict__ att_d,
                                               float* __restrict__ a_s,
                                               float* __restrict__ a_d, int N) {
    int tid = blockIdx.x * 256 + threadIdx.x;
    if (tid >= N * H4) return;
    int n = tid >> 2, hh = tid & 3;
    const float* row = h + (size_t)n * F + hh * C32;
    const float* ws = att_s + hh * C32;
    const float* wd = att_d + hh * C32;
    float s = 0.f, d = 0.f;
#pragma unroll 8
    for (int c = 0; c < C32; ++c) {
        s = fmaf(row[c], ws[c], s);
        d = fmaf(row[c], wd[c], d);
    }
    a_s[tid] = s;
    a_d[tid] = d;
}

// float -> order-preserving unsigned key
__device__ __forceinline__ unsigned f2key(float f) {
    unsigned u = __float_as_uint(f);
    return (u & 0x80000000u) ? ~u : (u | 0x80000000u);
}
__device__ __forceinline__ float key2f(unsigned k) {
    return (k & 0x80000000u) ? __uint_as_float(k & 0x7FFFFFFFu) : __uint_as_float(~k);
}

// ---------------------------------------------------------------- pass 1: e = lrelu(a_src[s]+a_dst[d]); segment max via atomicMax
__global__ __launch_bounds__(256) void gat_edge_e(const int* __restrict__ ei, int E, int N,
                                                  const float* __restrict__ a_s,
                                                  const float* __restrict__ a_d,
                                                  float* __restrict__ e_buf,
                                                  unsigned* __restrict__ emax) {
    size_t tid = (size_t)blockIdx.x * 256 + threadIdx.x;
    if (tid >= (size_t)(E + N) * H4) return;
    int i = (int)(tid >> 2), hh = (int)(tid & 3);
    int s, d;
    if (i < E) { s = ei[i]; d = ei[E + i]; } else { s = d = i - E; }
    float e = a_s[s * H4 + hh] + a_d[d * H4 + hh];
    e = (e > 0.f) ? e : NEG_SLOPE * e;
    e_buf[tid] = e;
    atomicMax(&emax[d * H4 + hh], f2key(e));
}

// ---------------------------------------------------------------- pass 2: exp(e - max) and segment sum
__global__ __launch_bounds__(256) void gat_edge_exp(const int* __restrict__ ei, int E, int N,
                                                    const unsigned* __restrict__ emax,
                                                    float* __restrict__ e_buf,
                                                    float* __restrict__ denom) {
    size_t tid = (size_t)blockIdx.x * 256 + threadIdx.x;
    if (tid >= (size_t)(E + N) * H4) return;
    int i = (int)(tid >> 2), hh = (int)(tid & 3);
    int d;
    if (i < E) { d = ei[E + i]; } else { d = i - E; }
    float m = key2f(emax[d * H4 + hh]);
    float ex = expf(e_buf[tid] - m);
    e_buf[tid] = ex;
    unsafeAtomicAdd(&denom[d * H4 + hh], ex);
}

// ---------------------------------------------------------------- pass 3: one wave32 per edge, lane = channel; scatter-add message
__global__ __launch_bounds__(256) void gat_edge_msg(const int* __restrict__ ei, int E, int N,
                                                    const float* __restrict__ e_buf,
                                                    const float* __restrict__ denom,
                                                    const float* __restrict__ hfeat,
                                                    float* __restrict__ agg) {
    int edge = blockIdx.x * 8 + (threadIdx.x >> 5);
    int lane = threadIdx.x & 31;
    if (edge >= E + N) return;
    int s, d;
    if (edge < E) { s = ei[edge]; d = ei[E + edge]; } else { s = d = edge - E; }
    const float* hs = hfeat + (size_t)s * F;
    float* ad = agg + (size_t)d * F;
#pragma unroll
    for (int j = 0; j < H4; ++j) {
        float alpha = e_buf[(size_t)edge * H4 + j] / denom[d * H4 + j];
        float val = hs[j * C32 + lane] * alpha;
        unsafeAtomicAdd(&ad[j * C32 + lane], val);
    }
}

// ---------------------------------------------------------------- bias + relu in place (layer-1 epilogue)
__global__ __launch_bounds__(256) void gat_bias_relu(float* __restrict__ a,
                                                     const float* __restrict__ b, size_t n) {
    size_t tid = (size_t)blockIdx.x * 256 + threadIdx.x;
    if (tid >= n) return;
    float v = a[tid] + b[tid & (F - 1)];
    a[tid] = v > 0.f ? v : 0.f;
}

// ---------------------------------------------------------------- layer-2 GEMM via V_WMMA_F32_16X16X4_F32
// C[N,128] = A[N,128] @ B[128,128]; block = 16-row tile, 8 waves cover 8 N-tiles.
__global__ __launch_bounds__(256) void gat_gemm_wmma(const float* __restrict__ A,
                                                     const float* __restrict__ B,
                                                     float* __restrict__ C, int N) {
    __shared__ float As[16 * F];  // 8 KB A tile
    int mbase = blockIdx.x * 16;
    int t = threadIdx.x;
    // cooperative A-tile load: 2048 floats as 512 float4
    for (int i = t; i < 512; i += 256) {
        int fbase = i * 4;
        int row = fbase >> 7;
        int col = fbase & (F - 1);
        float4 v = {0.f, 0.f, 0.f, 0.f};
        int gr = mbase + row;
        if (gr < N) v = *(const float4*)(A + (size_t)gr * F + col);
        *(float4*)(As + fbase) = v;
    }
    __syncthreads();

    int wave = t >> 5;       // n-tile 0..7
    int lane = t & 31;
    int half = lane >> 4;    // K half selector
    int l16 = lane & 15;     // M (for A) / N (for B,D)
    int ncol = wave * 16 + l16;

    v8f acc = {};
#pragma unroll
    for (int ks = 0; ks < 32; ++ks) {
        int k0 = ks * 4 + half * 2;
        v2f a;
        a.x = As[l16 * F + k0];
        a.y = As[l16 * F + k0 + 1];
        v2f b;
        b.x = B[(size_t)k0 * F + ncol];
        b.y = B[(size_t)(k0 + 1) * F + ncol];
        acc = __builtin_amdgcn_wmma_f32_16x16x4_f32(
            /*neg_a=*/false, a, /*neg_b=*/false, b,
            /*c_mod=*/(short)0, acc, /*reuse_a=*/false, /*reuse_b=*/false);
    }
#pragma unroll
    for (int r = 0; r < 8; ++r) {
        int row = mbase + r + half * 8;
        if (row < N) C[(size_t)row * F + ncol] = acc[r];
    }
}

// ---------------------------------------------------------------- final: mean heads + bias2, relu, @W_lin + b_lin (wave per node)
__global__ __launch_bounds__(256) void gat_final(const float* __restrict__ agg,
                                                 const float* __restrict__ bias2,
                                                 const float* __restrict__ wlin,
                                                 const float* __restrict__ blin,
                                                 float* __restrict__ out, int N) {
    int wid = (int)(((size_t)blockIdx.x * 256 + threadIdx.x) >> 5);
    int lane = threadIdx.x & 31;
    if (wid >= N) return;
    const float* row = agg + (size_t)wid * F;
    float v = 0.25f * (row[lane] + row[lane + 32] + row[lane + 64] + row[lane + 96]) + bias2[lane];
    v = v > 0.f ? v : 0.f;
    float p = v * wlin[lane];
#pragma unroll
    for (int off = 16; off; off >>= 1) p += __shfl_xor(p, off, 32);
    if (lane == 0) out[wid] = p + blin[0];
}

static inline size_t align256(size_t x) { return (x + 255) & ~(size_t)255; }

extern "C" void kernel_launch(void* const* d_in, const int* in_sizes, int n_in,
                              void* d_out, int out_size, void* d_ws, size_t ws_size,
                              hipStream_t stream) {
    const float* x        = (const float*)d_in[0];
    const int*   ei       = (const int*)d_in[1];
    const float* W1       = (const float*)d_in[2];
    const float* att_src1 = (const float*)d_in[3];
    const float* att_dst1 = (const float*)d_in[4];
    const float* bias1    = (const float*)d_in[5];
    const float* W2       = (const float*)d_in[6];
    const float* att_src2 = (const float*)d_in[7];
    const float* att_dst2 = (const float*)d_in[8];
    const float* bias2    = (const float*)d_in[9];
    const float* W_lin    = (const float*)d_in[10];
    const float* b_lin    = (const float*)d_in[11];
    float* out = (float*)d_out;

    const int N = in_sizes[0] / 8;
    const int E = in_sizes[1] / 2;
    const int EN = E + N;

    // workspace layout
    char* ws = (char*)d_ws;
    size_t off = 0;
    float* h     = (float*)(ws + off); off += align256((size_t)N * F * 4);   // features (h1, then h2)
    float* agg   = (float*)(ws + off); off += align256((size_t)N * F * 4);   // aggregation target
    float* e_buf = (float*)(ws + off); off += align256((size_t)EN * H4 * 4); // per-edge logits/weights
    float* a_s   = (float*)(ws + off); off += align256((size_t)N * H4 * 4);
    float* a_d   = (float*)(ws + off); off += align256((size_t)N * H4 * 4);
    unsigned* emax = (unsigned*)(ws + off); off += align256((size_t)N * H4 * 4);
    float* denom = (float*)(ws + off); off += align256((size_t)N * H4 * 4);

    const unsigned KEY_NEG_INF = 0x007FFFFFu; // f2key(-inf)
    const size_t nNH = (size_t)N * H4;
    const size_t nNF = (size_t)N * F;
    const size_t nEH = (size_t)EN * H4;

    dim3 blk(256);
    int gNH  = (int)((nNH + 255) / 256);
    int gNF  = (int)((nNF + 255) / 256);
    int gEH  = (int)((nEH + 255) / 256);
    int gMsg = (EN + 7) / 8;
    int gM16 = (N + 15) / 16;
    int gFin = (int)(((size_t)N * 32 + 255) / 256);

    // ---------------- layer 1 ----------------
    gat_fill_u32<<<gNH, blk, 0, stream>>>(emax, KEY_NEG_INF, nNH);
    gat_fill_u32<<<gNH, blk, 0, stream>>>((unsigned*)denom, 0u, nNH);
    gat_fill_u32<<<gNF, blk, 0, stream>>>((unsigned*)agg, 0u, nNF);

    gat_lin1<<<gNF, blk, 0, stream>>>(x, W1, h, N);
    gat_att<<<gNH, blk, 0, stream>>>(h, att_src1, att_dst1, a_s, a_d, N);
    gat_edge_e<<<gEH, blk, 0, stream>>>(ei, E, N, a_s, a_d, e_buf, emax);
    gat_edge_exp<<<gEH, blk, 0, stream>>>(ei, E, N, emax, e_buf, denom);
    gat_edge_msg<<<gMsg, blk, 0, stream>>>(ei, E, N, e_buf, denom, h, agg);
    gat_bias_relu<<<gNF, blk, 0, stream>>>(agg, bias1, nNF);

    // ---------------- layer 2 ----------------
    gat_gemm_wmma<<<gM16, blk, 0, stream>>>(agg, W2, h, N);  // h = relu(agg1) @ W2
    gat_att<<<gNH, blk, 0, stream>>>(h, att_src2, att_dst2, a_s, a_d, N);

    gat_fill_u32<<<gNH, blk, 0, stream>>>(emax, KEY_NEG_INF, nNH);
    gat_fill_u32<<<gNH, blk, 0, stream>>>((unsigned*)denom, 0u, nNH);
    gat_fill_u32<<<gNF, blk, 0, stream>>>((unsigned*)agg, 0u, nNF);

    gat_edge_e<<<gEH, blk, 0, stream>>>(ei, E, N, a_s, a_d, e_buf, emax);
    gat_edge_exp<<<gEH, blk, 0, stream>>>(ei, E, N, emax, e_buf, denom);
    gat_edge_msg<<<gMsg, blk, 0, stream>>>(ei, E, N, e_buf, denom, h, agg);

    gat_final<<<gFin, blk, 0, stream>>>(agg, bias2, W_lin, b_lin, out, N);
}